// CoucheinitialeGNN_90082644066770
// MI455X (gfx1250) — compile-verified
//
#include <hip/hip_runtime.h>

#define N_NODES   100000
#define N_EDGES   2000000
#define EDGE_HID  64
#define EDGE_OUT  10
#define HIDDEN    20
#define THRESHOLD 10.0f
#define TILES_PER_WAVE 5   // 125000 tiles / 5 = 25000 waves = 3125 blocks exactly

// N_EDGES must tile exactly: every tile < ntiles has 16 valid edges,
// so no per-lane bounds guards are needed (removes exec-mask churn).
static_assert(N_EDGES % 16 == 0, "edge tiling must be exact");

typedef __attribute__((ext_vector_type(2))) float v2f;
typedef __attribute__((ext_vector_type(8))) float v8f;

// ---------------------------------------------------------------------------
// K1: zero the output buffer (out[:,0,:] doubles as sum_w scratch,
//     out[:,1,:] accumulates sum_features)
// ---------------------------------------------------------------------------
__global__ void k_zero(float* __restrict__ out, int n4) {
    int i = blockIdx.x * blockDim.x + threadIdx.x;
    int stride = gridDim.x * blockDim.x;
    float4 z = make_float4(0.f, 0.f, 0.f, 0.f);
    for (; i < n4; i += stride) ((float4*)out)[i] = z;
}

// ---------------------------------------------------------------------------
// Per-wave weight preload into registers for the WMMA edge MLP.
// A layout (16x4 f32): VGPR0 holds K=2*half, VGPR1 holds K=2*half+1.
// B layout (4x16 f32): mirrored; lane col = lane&15 holds W2[k][col].
// ---------------------------------------------------------------------------
__device__ __forceinline__ void load_weights(
    const float* __restrict__ W1, const float* __restrict__ b1,
    const float* __restrict__ W2, int half, int col,
    float* w1lo, float* w1hi, float* b1lo, float* b1hi, v2f* Bt)
{
#pragma unroll
    for (int t = 0; t < 16; ++t) {
        int k0 = 4 * t + 2 * half;
        w1lo[t] = W1[k0];
        w1hi[t] = W1[k0 + 1];
        b1lo[t] = b1[k0];
        b1hi[t] = b1[k0 + 1];
        float bx = 0.f, by = 0.f;
        if (col < EDGE_OUT) {
            bx = W2[k0 * EDGE_OUT + col];
            by = W2[(k0 + 1) * EDGE_OUT + col];
        }
        v2f bv; bv.x = bx; bv.y = by;
        Bt[t] = bv;
    }
}

// hidden = relu(d*W1+b1); C(16x16) = hidden(16x64) @ W2pad(64x16) via 16 WMMAs
__device__ __forceinline__ v8f edge_mlp_tile(
    float d, const float* w1lo, const float* w1hi,
    const float* b1lo, const float* b1hi, const v2f* Bt)
{
    v8f c = {};
#pragma unroll
    for (int t = 0; t < 16; ++t) {
        v2f A;
        A.x = fmaxf(fmaf(d, w1lo[t], b1lo[t]), 0.0f);
        A.y = fmaxf(fmaf(d, w1hi[t], b1hi[t]), 0.0f);
        c = __builtin_amdgcn_wmma_f32_16x16x4_f32(
            /*neg_a=*/false, A, /*neg_b=*/false, Bt[t],
            /*c_mod=*/(short)0, c, /*reuse_a=*/false, /*reuse_b=*/false);
    }
    return c;
}

// ---------------------------------------------------------------------------
// K2: pass 1 — build sum_w (into out[n,0,:]) via atomics.
//     One wave handles 16 edges per tile; C row m=8*half+r, col=lane&15.
// ---------------------------------------------------------------------------
__global__ void k_edge_pass1(
    const float* __restrict__ edge_attr, const int* __restrict__ src,
    const float* __restrict__ W1, const float* __restrict__ b1,
    const float* __restrict__ W2, const float* __restrict__ b2,
    float* __restrict__ out, int ntiles)
{
    int lane = threadIdx.x & 31;
    int col  = lane & 15;
    int half = lane >> 4;
    int wave = (blockIdx.x * blockDim.x + threadIdx.x) >> 5;

    float w1lo[16], w1hi[16], b1lo[16], b1hi[16];
    v2f   Bt[16];
    load_weights(W1, b1, W2, half, col, w1lo, w1hi, b1lo, b1hi, Bt);
    float b2c = (col < EDGE_OUT) ? b2[col] : 0.f;

    for (int i = 0; i < TILES_PER_WAVE; ++i) {
        int tile = wave * TILES_PER_WAVE + i;     // wave-uniform
        if (tile >= ntiles) break;                // uniform break: EXEC stays full
        int base   = tile * 16;
        int e_lane = base + col;
        if (tile + 1 < ntiles) {                  // uniform: prefetch next tile
            __builtin_prefetch(&edge_attr[e_lane + 16], 0, 1);
            __builtin_prefetch(&src[e_lane + 16], 0, 1);
        }
        float d    = edge_attr[e_lane];
        int s_reg  = src[e_lane];
        int bu_reg = min(max((int)(d * (10.0f / THRESHOLD)), 0), 9);

        v8f c = edge_mlp_tile(d, w1lo, w1hi, b1lo, b1hi, Bt);

#pragma unroll
        for (int r = 0; r < 8; ++r) {
            int m  = 8 * half + r;
            int s  = __shfl(s_reg, m, 32);
            int bu = __shfl(bu_reg, m, 32);
            if (col < EDGE_OUT)
                atomicAdd(&out[s * 40 + 10 + col], c[r] + b2c);  // mlp dims
            if (col == bu)
                atomicAdd(&out[s * 40 + col], 1.0f);             // one-hot dim
        }
    }
}

// ---------------------------------------------------------------------------
// K3: pass 2 — recompute eac (cheaper than spilling 160MB), normalize by
//     sum_w[src], scale by rho, accumulate sum_features into out[n,1,:].
// ---------------------------------------------------------------------------
__global__ void k_edge_pass2(
    const float* __restrict__ edge_attr, const int* __restrict__ src,
    const int* __restrict__ dst, const float* __restrict__ x,
    const float* __restrict__ a_p, const float* __restrict__ b_p,
    const float* __restrict__ W1, const float* __restrict__ b1,
    const float* __restrict__ W2, const float* __restrict__ b2,
    float* __restrict__ out, int ntiles)
{
    int lane = threadIdx.x & 31;
    int col  = lane & 15;
    int half = lane >> 4;
    int wave = (blockIdx.x * blockDim.x + threadIdx.x) >> 5;

    float a0   = a_p[0];
    float bexp = b_p[0];

    float w1lo[16], w1hi[16], b1lo[16], b1hi[16];
    v2f   Bt[16];
    load_weights(W1, b1, W2, half, col, w1lo, w1hi, b1lo, b1hi, Bt);
    float b2c = (col < EDGE_OUT) ? b2[col] : 0.f;

    for (int i = 0; i < TILES_PER_WAVE; ++i) {
        int tile = wave * TILES_PER_WAVE + i;
        if (tile >= ntiles) break;
        int base   = tile * 16;
        int e_lane = base + col;
        if (tile + 1 < ntiles) {
            __builtin_prefetch(&edge_attr[e_lane + 16], 0, 1);
            __builtin_prefetch(&src[e_lane + 16], 0, 1);
            __builtin_prefetch(&dst[e_lane + 16], 0, 1);
        }
        float d    = edge_attr[e_lane];
        int s_reg  = src[e_lane];
        int d_reg  = dst[e_lane];
        int bu_reg = min(max((int)(d * (10.0f / THRESHOLD)), 0), 9);
        float diff = fabsf(a0 * x[s_reg] - (1.0f - a0) * x[d_reg]);
        float rho_reg = powf(diff, bexp);

        v8f c = edge_mlp_tile(d, w1lo, w1hi, b1lo, b1hi, Bt);

#pragma unroll
        for (int r = 0; r < 8; ++r) {
            int m     = 8 * half + r;
            int s     = __shfl(s_reg, m, 32);
            int bu    = __shfl(bu_reg, m, 32);
            float rho = __shfl(rho_reg, m, 32);
            if (col < EDGE_OUT) {
                float mlp  = c[r] + b2c;
                float swm  = out[s * 40 + 10 + col];
                float wt_m = (swm != 0.f) ? (mlp / swm) : 0.01f;
                float oh   = (col == bu) ? 1.f : 0.f;
                float swo  = out[s * 40 + col];
                float wt_o = (swo != 0.f) ? (oh / swo) : 0.01f;
                atomicAdd(&out[s * 40 + 20 + col], rho * wt_o);   // dims 0..9
                atomicAdd(&out[s * 40 + 30 + col], rho * wt_m);   // dims 10..19
            }
        }
    }
}

// ---------------------------------------------------------------------------
// K4: node update — h = sigmoid(x*g1 + sf@g2^T + bias) overwrites out[n,0,:]
// ---------------------------------------------------------------------------
__global__ void k_node(
    const float* __restrict__ x, const float* __restrict__ gamma1,
    const float* __restrict__ gamma2, const float* __restrict__ bias,
    float* __restrict__ out)
{
    __shared__ float g2[HIDDEN * HIDDEN];
    __shared__ float g1s[HIDDEN];
    __shared__ float bs[HIDDEN];
    for (int i = threadIdx.x; i < HIDDEN * HIDDEN; i += blockDim.x) g2[i] = gamma2[i];
    if (threadIdx.x < HIDDEN) {
        g1s[threadIdx.x] = gamma1[threadIdx.x];
        bs[threadIdx.x]  = bias[threadIdx.x];
    }
    __syncthreads();

    int n = blockIdx.x * blockDim.x + threadIdx.x;
    if (n >= N_NODES) return;

    float xv = x[n];
    float sf[HIDDEN];
#pragma unroll
    for (int j = 0; j < HIDDEN; ++j) sf[j] = out[n * 40 + 20 + j];
#pragma unroll
    for (int i = 0; i < HIDDEN; ++i) {
        float acc = fmaf(g1s[i], xv, bs[i]);
#pragma unroll
        for (int j = 0; j < HIDDEN; ++j) acc = fmaf(g2[i * HIDDEN + j], sf[j], acc);
        out[n * 40 + i] = 1.0f / (1.0f + expf(-acc));
    }
}

// ---------------------------------------------------------------------------
extern "C" void kernel_launch(void* const* d_in, const int* in_sizes, int n_in,
                              void* d_out, int out_size, void* d_ws, size_t ws_size,
                              hipStream_t stream) {
    const float* x         = (const float*)d_in[0];
    const float* edge_attr = (const float*)d_in[1];
    const float* a_p       = (const float*)d_in[2];
    const float* b_p       = (const float*)d_in[3];
    const float* gamma1    = (const float*)d_in[4];
    const float* gamma2    = (const float*)d_in[5];
    const float* bias      = (const float*)d_in[6];
    const float* W1        = (const float*)d_in[7];
    const float* b1        = (const float*)d_in[8];
    const float* W2        = (const float*)d_in[9];
    const float* b2        = (const float*)d_in[10];
    const int*   eidx      = (const int*)d_in[11];
    const int*   src       = eidx;
    const int*   dst       = eidx + N_EDGES;
    float*       out       = (float*)d_out;

    // K1: zero out (N*2*20 floats, divisible by 4)
    int n4 = (N_NODES * 2 * HIDDEN) / 4;
    k_zero<<<2048, 256, 0, stream>>>(out, n4);

    int ntiles = N_EDGES / 16;                                    // 125000
    int waves  = (ntiles + TILES_PER_WAVE - 1) / TILES_PER_WAVE;  // 25000
    int blocks = (waves * 32 + 255) / 256;                        // 3125 exactly

    k_edge_pass1<<<blocks, 256, 0, stream>>>(edge_attr, src, W1, b1, W2, b2,
                                             out, ntiles);
    k_edge_pass2<<<blocks, 256, 0, stream>>>(edge_attr, src, dst, x, a_p, b_p,
                                             W1, b1, W2, b2, out, ntiles);
    k_node<<<(N_NODES + 255) / 256, 256, 0, stream>>>(x, gamma1, gamma2, bias, out);
}